// RadiusInteractionGraph_73246372266582
// MI455X (gfx1250) — compile-verified
//
#include <hip/hip_runtime.h>
#include <math.h>

// Problem constants (match reference)
#define B_MOL   128
#define NPM     512
#define KNN     32
#define CUT2    100.0f      // CUTOFF^2
#define BIG     1e10f

typedef __attribute__((ext_vector_type(2))) float v2f;
typedef __attribute__((ext_vector_type(8))) float v8f;

// One block = one (molecule, 16-row strip). 512 threads = 16 wave32s.
// LDS: 512 atoms as float4(x,y,z,|p|^2) = 8KB, plus 16x512 d2 strip = 32KB.
__global__ __launch_bounds__(512)
void radius_graph_wmma_kernel(const float* __restrict__ pos,
                              float* __restrict__ out)
{
    __shared__ float4 spos[NPM];
    __shared__ float  sd2[16 * NPM];

    const int tid   = threadIdx.x;
    const int lane  = tid & 31;
    const int wave  = tid >> 5;
    const int blk   = blockIdx.x;
    const int mol   = blk >> 5;      // 32 strips per molecule
    const int strip = blk & 31;
    const int molbase = mol * NPM;

    // ---- Phase 1: stage molecule positions + |p|^2 into LDS ----
    {
        const float* p = pos + (size_t)(molbase + tid) * 3;
        float x = p[0], y = p[1], z = p[2];
        float4 v; v.x = x; v.y = y; v.z = z; v.w = x*x + y*y + z*z;
        spos[tid] = v;
    }
    __syncthreads();

    // ---- Phase 2: WMMA Gram tiles -> masked squared distances in LDS ----
    // A (16x4 f32): lanes 0-15 hold K=0,1 (x,y); lanes 16-31 hold K=2,3 (z,0).
    {
        const int rsel = lane & 15;
        const int hi   = lane >> 4;
        const float4 pa = spos[strip * 16 + rsel];
        v2f a;
        a.x = hi ? pa.z : pa.x;
        a.y = hi ? 0.0f : pa.y;
        // each wave computes 2 of the 32 column tiles
        #pragma unroll
        for (int tt = 0; tt < 2; ++tt) {
            const int t = wave * 2 + tt;
            const float4 pb = spos[t * 16 + rsel];
            v2f b;
            b.x = hi ? pb.z : pb.x;
            b.y = hi ? 0.0f : pb.y;
            v8f c = {};
            // D = A(16x4) x B(4x16) : one f32 WMMA per tile
            c = __builtin_amdgcn_wmma_f32_16x16x4_f32(
                    false, a, false, b, (short)0, c, false, false);
            const int n = t * 16 + rsel;            // local column atom
            #pragma unroll
            for (int r = 0; r < 8; ++r) {
                const int m   = r + hi * 8;         // row within strip
                const float sqm = spos[strip * 16 + m].w;
                float d2 = sqm + pb.w - 2.0f * c[r];
                d2 = fmaxf(d2, 0.0f);
                if ((strip * 16 + m) == n || d2 > CUT2) d2 = BIG;
                sd2[m * NPM + n] = d2;
            }
        }
    }
    __syncthreads();

    // ---- Phase 3: wave w extracts 32 nearest neighbors of strip row w ----
    {
        const int row  = wave;                          // 0..15
        const int rowg = molbase + strip * 16 + row;    // global center atom
        float v[16]; int ix[16];
        #pragma unroll
        for (int k = 0; k < 16; ++k) {
            v[k]  = sd2[row * NPM + lane + 32 * k];     // conflict-free stride
            ix[k] = lane + 32 * k;
        }
        float* osrc = out;
        float* odst = out + (size_t)B_MOL * NPM * KNN;
        float* ow   = out + (size_t)2 * B_MOL * NPM * KNN;

        for (int it = 0; it < KNN; ++it) {
            // local min over this lane's 16 values (tie -> lower index)
            float mv = v[0]; int mi = ix[0];
            #pragma unroll
            for (int k = 1; k < 16; ++k)
                if (v[k] < mv || (v[k] == mv && ix[k] < mi)) { mv = v[k]; mi = ix[k]; }
            // wave32 xor-reduction
            #pragma unroll
            for (int off = 16; off > 0; off >>= 1) {
                float ov = __shfl_xor(mv, off, 32);
                int   oi = __shfl_xor(mi, off, 32);
                if (ov < mv || (ov == mv && oi < mi)) { mv = ov; mi = oi; }
            }
            const bool valid = mv < 0.5f * BIG;
            // retire the winning element in its owner lane
            if (lane == (mi & 31)) v[mi >> 5] = 2.0f * BIG;
            if (lane == 0) {
                const size_t e = (size_t)rowg * KNN + it;
                osrc[e] = (float)(valid ? (molbase + mi) : rowg);
                odst[e] = (float)rowg;
                ow[e]   = valid ? sqrtf(fmaxf(mv, 1e-12f)) : 0.0f;
            }
        }
    }
}

extern "C" void kernel_launch(void* const* d_in, const int* in_sizes, int n_in,
                              void* d_out, int out_size, void* d_ws, size_t ws_size,
                              hipStream_t stream)
{
    (void)in_sizes; (void)n_in; (void)d_ws; (void)ws_size; (void)out_size;
    const float* pos = (const float*)d_in[0];   // [N,3] float32
    // d_in[1] (batch) is implied by the uniform block structure; unused.
    float* out = (float*)d_out;                 // [src | dst | weight], each B*NPM*K

    dim3 grid(B_MOL * (NPM / 16));              // 4096 blocks
    dim3 block(512);                            // 16 wave32s
    radius_graph_wmma_kernel<<<grid, block, 0, stream>>>(pos, out);
}